// ConsistencyLoss_16501264351800
// MI455X (gfx1250) — compile-verified
//
#include <hip/hip_runtime.h>

// ---------------- Problem constants (from reference) ----------------
#define A_N 49104
#define B_N 8
#define C_N 20
#define M_N 32
#define NN_N 16
#define AC (A_N * C_N)   // 982080
#define CHUNKS 192       // ceil(A_N / 256)

// ---------------- Workspace layout (float offsets) ------------------
// alpha[A*C] | keep[B*M] | has[B] | pad | P[B][CHUNKS][4]   (~3.96 MB)
#define ALPHA_OFF 0
#define KEEP_OFF  AC                 // 982080
#define HAS_OFF   (KEEP_OFF + 256)   // 982336
#define P_OFF     982352             // 16B aligned; 6144 floats

typedef __attribute__((ext_vector_type(2))) float v2f;
typedef __attribute__((ext_vector_type(8))) float v8f;

// ---------------- Kernel 0: keep/has masks ---------------------------
__global__ void prep_kernel(const float* __restrict__ ema_classes,
                            const int* __restrict__ ema_counts,
                            const float* __restrict__ annotations,
                            float* __restrict__ ws) {
    int t = threadIdx.x;          // 256 threads = B*M pairs
    int b = t >> 5, m = t & 31;
    float cm = ema_classes[b * M_N + m];
    bool member = false;
#pragma unroll
    for (int n = 0; n < NN_N; ++n)
        member = member || (annotations[(b * NN_N + n) * 5 + 4] == cm);
    bool valid = m < ema_counts[b];
    float k = (valid && member) ? 1.0f : 0.0f;
    ws[KEEP_OFF + t] = k;

    __shared__ float sk[256];
    sk[t] = k;
    __syncthreads();
    if (t < B_N) {
        float h = 0.0f;
#pragma unroll
        for (int mm = 0; mm < M_N; ++mm) h = fmaxf(h, sk[t * M_N + mm]);
        ws[HAS_OFF + t] = h;
    }
}

// ---------------- Kernel 1: alpha_sum[a,c] (batch-sum, coalesced) ----
__global__ void alpha_kernel(const float* __restrict__ ecls,
                             float* __restrict__ ws) {
    int i = blockIdx.x * 256 + threadIdx.x;
    if (i >= AC) return;
    float s = 0.0f;
#pragma unroll
    for (int b = 0; b < B_N; ++b) s += ecls[(size_t)b * AC + i];
    // sum_b (a0 + e*(a1-a0)) = B*a0 + (a1-a0)*sum_b e = 0.4 + 0.9*s
    ws[ALPHA_OFF + i] = 0.4f + 0.9f * s;
}

// ---------------- per-class focal BCE term ---------------------------
__device__ __forceinline__ float focal_bce(float c, float e, float al) {
    const float eps = 1e-4f;
    float p = fminf(fmaxf(c, eps), 1.0f - eps);
    float q = fminf(fmaxf(e, eps), 1.0f - eps);
    float bce = -(q * logf(p) + (1.0f - q) * logf(1.0f - p));
    float d = q - p;                 // gamma == 2.0  ->  |d|^2 == d*d
    return al * d * d * bce;
}

// ---------------- Kernel 2: main streaming pass ----------------------
__global__ __launch_bounds__(256) void main_kernel(
    const float* __restrict__ cls, const float* __restrict__ reg,
    const float* __restrict__ anchors, const float* __restrict__ ecls,
    const float* __restrict__ ema_bboxes, float* __restrict__ ws) {
    const int tid = threadIdx.x;
    const int chunk = blockIdx.x;   // anchor chunk
    const int b = blockIdx.y;       // batch index

    __shared__ float4 sbox[M_N];
    __shared__ float sarea[M_N];
    __shared__ float skeep[M_N];
    __shared__ float shas;
    __shared__ float red[8][4];

    if (tid < M_N) {
        float4 bx = ((const float4*)ema_bboxes)[b * M_N + tid];
        sbox[tid] = bx;
        sarea[tid] = (bx.z - bx.x) * (bx.w - bx.y);
        skeep[tid] = ws[KEEP_OFF + b * M_N + tid];
    }
    if (tid == 0) shas = ws[HAS_OFF + b];
    __syncthreads();

    int a = chunk * 256 + tid;
    float clsv = 0.0f, regv = 0.0f, nposv = 0.0f;
    if (a < A_N) {
        float4 an = ((const float4*)anchors)[a];
        float aw = an.z - an.x, ah = an.w - an.y;
        float acx = an.x + 0.5f * aw, acy = an.y + 0.5f * ah;
        float area_a = aw * ah;

        float best = -1.0f; int arg = 0;
#pragma unroll 8
        for (int m = 0; m < M_N; ++m) {
            float4 bx = sbox[m];
            float iw = fminf(an.z, bx.z) - fmaxf(an.x, bx.x);
            float ih = fminf(an.w, bx.w) - fmaxf(an.y, bx.y);
            iw = fmaxf(iw, 0.0f); ih = fmaxf(ih, 0.0f);
            float inter = iw * ih;
            float uni = fmaxf(area_a + sarea[m] - inter, 1e-8f);
            float iou = inter / uni;
            float v = (skeep[m] != 0.0f) ? iou : -1.0f;
            if (v > best) { best = v; arg = m; }   // first-max == jnp.argmax
        }
        bool hasb = (shas != 0.0f);
        bool pos = (best >= 0.5f) && hasb;
        bool assigned = ((best < 0.4f) || (best >= 0.5f)) && hasb;
        nposv = pos ? 1.0f : 0.0f;

        if (pos) {
            float4 ab = sbox[arg];
            float gw = ab.z - ab.x, gh = ab.w - ab.y;
            float gcx = ab.x + 0.5f * gw, gcy = ab.y + 0.5f * gh; // unclipped
            gw = fmaxf(gw, 1.0f); gh = fmaxf(gh, 1.0f);
            float dx = ((gcx - acx) / aw) / 0.1f;
            float dy = ((gcy - acy) / ah) / 0.1f;
            float dw = logf(gw / aw) / 0.2f;
            float dh = logf(gh / ah) / 0.2f;
            float4 rg = ((const float4*)reg)[(size_t)b * A_N + a];
            float t4[4] = {dx - rg.x, dy - rg.y, dw - rg.z, dh - rg.w};
#pragma unroll
            for (int k = 0; k < 4; ++k) {
                float d = fabsf(t4[k]);
                regv += (d <= (1.0f / 9.0f)) ? 4.5f * d * d : d - (0.5f / 9.0f);
            }
        }

        if (assigned) {
            const float4* c4 = (const float4*)(cls  + ((size_t)b * A_N + a) * C_N);
            const float4* e4 = (const float4*)(ecls + ((size_t)b * A_N + a) * C_N);
            const float4* a4 = (const float4*)(ws + ALPHA_OFF + (size_t)a * C_N);
#pragma unroll
            for (int j = 0; j < 5; ++j) {
                float4 pc = c4[j], pe = e4[j], pa = a4[j];
                clsv += focal_bce(pc.x, pe.x, pa.x);
                clsv += focal_bce(pc.y, pe.y, pa.y);
                clsv += focal_bce(pc.z, pe.z, pa.z);
                clsv += focal_bce(pc.w, pe.w, pa.w);
            }
        }
    }

    // wave32 reduction, then cross-wave via LDS
#pragma unroll
    for (int off = 16; off > 0; off >>= 1) {
        clsv += __shfl_xor(clsv, off, 32);
        regv += __shfl_xor(regv, off, 32);
        nposv += __shfl_xor(nposv, off, 32);
    }
    if ((tid & 31) == 0) {
        int w = tid >> 5;
        red[w][0] = clsv; red[w][1] = regv; red[w][2] = nposv;
    }
    __syncthreads();
    if (tid == 0) {
        float s0 = 0, s1 = 0, s2 = 0;
#pragma unroll
        for (int w = 0; w < 8; ++w) { s0 += red[w][0]; s1 += red[w][1]; s2 += red[w][2]; }
        float* P = ws + P_OFF + ((size_t)b * CHUNKS + chunk) * 4;
        P[0] = s0; P[1] = s1; P[2] = s2; P[3] = 0.0f;
    }
}

// ---------------- Kernel 3: WMMA chunk reduction + final scalars -----
__global__ void finalize_kernel(const float* __restrict__ ws,
                                float* __restrict__ out) {
    const int tid = threadIdx.x;   // one wave32
    const float* P = ws + P_OFF;
    __shared__ float nposs[8];
    __shared__ float contrib[16];

    if (tid < 8) {
        float s = 0.0f;
        for (int k = 0; k < CHUNKS; ++k)
            s += P[((size_t)tid * CHUNKS + k) * 4 + 2];
        nposs[tid] = s;             // exact: integer counts < 2^24
    }
    __syncthreads();

    // ones(16x4) x B(4x16), accumulated over 48 K-tiles == column sums of P.
    // Column n = (b = n&7, v = n>>3): v0 = cls partial, v1 = reg partial.
    // f32 B layout: VGPR j holds B[2h+j][n], lane = h*16 + n.
    const int n = tid & 15, h = tid >> 4;
    const int bb = n & 7, v = n >> 3;
    v2f av; av[0] = 1.0f; av[1] = 1.0f;
    v8f c = {};
    for (int it = 0; it < CHUNKS / 4; ++it) {
        int k0 = it * 4 + 2 * h;
        v2f bv;
        bv[0] = P[((size_t)bb * CHUNKS + k0) * 4 + v];
        bv[1] = P[((size_t)bb * CHUNKS + k0 + 1) * 4 + v];
        c = __builtin_amdgcn_wmma_f32_16x16x4_f32(false, av, false, bv,
                                                  (short)0, c, false, false);
    }
    float S = c[0];   // lanes 0..15 hold D[0][n] = total for column n

    if (tid < 16) {
        float np = nposs[n & 7];
        float mx = fmaxf(np, 1.0f);
        contrib[tid] = (tid < 8) ? (S / mx)
                                 : ((np > 0.0f) ? S / (4.0f * mx) : 0.0f);
    }
    __syncthreads();
    if (tid == 0) {
        float c0 = 0.0f, c1 = 0.0f;
#pragma unroll
        for (int i = 0; i < 8; ++i) { c0 += contrib[i]; c1 += contrib[8 + i]; }
        out[0] = c0 * 0.125f;   // cls_loss.mean()
        out[1] = c1 * 0.125f;   // reg_loss.mean()
    }
}

extern "C" void kernel_launch(void* const* d_in, const int* in_sizes, int n_in,
                              void* d_out, int out_size, void* d_ws, size_t ws_size,
                              hipStream_t stream) {
    const float* cls   = (const float*)d_in[0];
    const float* reg   = (const float*)d_in[1];
    const float* anch  = (const float*)d_in[2];
    const float* ecls  = (const float*)d_in[3];
    const float* eclas = (const float*)d_in[4];
    const float* ebox  = (const float*)d_in[5];
    const int*   ecnt  = (const int*)d_in[6];
    const float* ann   = (const float*)d_in[7];
    float* ws  = (float*)d_ws;
    float* out = (float*)d_out;

    prep_kernel<<<1, 256, 0, stream>>>(eclas, ecnt, ann, ws);
    alpha_kernel<<<(AC + 255) / 256, 256, 0, stream>>>(ecls, ws);
    main_kernel<<<dim3(CHUNKS, B_N), 256, 0, stream>>>(cls, reg, anch, ecls, ebox, ws);
    finalize_kernel<<<1, 32, 0, stream>>>(ws, out);
}